// TopKSparsifier_26611617366613
// MI455X (gfx1250) — compile-verified
//
#include <hip/hip_runtime.h>

#define ROW_D 4096
#define NT 256
#define QUARTER (ROW_D / 4)   // 1024

#if defined(__has_builtin)
#if __has_builtin(__builtin_amdgcn_global_load_async_to_lds_b128)
#define USE_ASYNC_LDS 1
#endif
#endif

typedef int v4i __attribute__((vector_size(4 * sizeof(int))));
typedef __attribute__((address_space(1))) v4i* g_v4i_ptr;  // global (AS1)
typedef __attribute__((address_space(3))) v4i* l_v4i_ptr;  // LDS (AS3)

// Find smallest bin such that cumulative histogram count >= k.
// hist has NT*BPT bins. Writes bin -> s_res[0], exclusive-prefix-count -> s_res[1].
template <unsigned BPT>
__device__ __forceinline__ void find_kth_bin(const unsigned* __restrict__ hist,
                                             unsigned k,
                                             unsigned* __restrict__ s_scan,
                                             unsigned* __restrict__ s_res) {
  const unsigned t = threadIdx.x;
  const unsigned base_bin = t * BPT;
  unsigned s = 0;
#pragma unroll
  for (unsigned j = 0; j < BPT; ++j) s += hist[base_bin + j];
  s_scan[t] = s;
  __syncthreads();
  // Hillis-Steele inclusive scan over 256 per-thread sums
#pragma unroll
  for (unsigned off = 1; off < NT; off <<= 1) {
    unsigned v = s_scan[t];
    unsigned a = (t >= off) ? s_scan[t - off] : 0u;
    __syncthreads();
    s_scan[t] = v + a;
    __syncthreads();
  }
  const unsigned incl = s_scan[t];
  const unsigned excl = incl - s;
  if (excl < k && incl >= k) {   // exactly one thread matches
    unsigned cum = excl;
#pragma unroll
    for (unsigned j = 0; j < BPT; ++j) {
      unsigned c = hist[base_bin + j];
      if (cum + c >= k) { s_res[0] = base_bin + j; s_res[1] = cum; break; }
      cum += c;
    }
  }
  __syncthreads();
}

__global__ void __launch_bounds__(NT)
topk_sparsify_26611617366613(const float* __restrict__ x,
                             float* __restrict__ out_val,
                             float* __restrict__ out_mask,
                             unsigned num_keep) {
  __shared__ float    s_row[ROW_D];     // 16 KB staged row
  __shared__ unsigned s_hist[2048];     // 8 KB radix histogram
  __shared__ unsigned s_scan[NT];
  __shared__ unsigned s_res[2];

  const unsigned t = threadIdx.x;
  const size_t row_base = (size_t)blockIdx.x * ROW_D;
  const float* xr = x + row_base;

  // ---- stage row into LDS: 4 coalesced b128 chunks per thread ----
#if USE_ASYNC_LDS
#pragma unroll
  for (int c = 0; c < 4; ++c) {
    const unsigned e = (unsigned)c * QUARTER + t * 4u;
    __builtin_amdgcn_global_load_async_to_lds_b128(
        (g_v4i_ptr)(xr + e), (l_v4i_ptr)(&s_row[e]), 0, 0);
  }
#else
#pragma unroll
  for (int c = 0; c < 4; ++c) {
    const unsigned e = (unsigned)c * QUARTER + t * 4u;
    *(float4*)(&s_row[e]) = *(const float4*)(xr + e);
  }
#endif

  // zero pass-1 histogram while the copy is in flight
#pragma unroll
  for (int j = 0; j < 8; ++j) s_hist[t * 8u + j] = 0u;

#if USE_ASYNC_LDS
#if defined(__has_builtin) && __has_builtin(__builtin_amdgcn_s_wait_asynccnt)
  __builtin_amdgcn_s_wait_asynccnt(0);
#else
  asm volatile("s_wait_asynccnt 0" ::: "memory");
#endif
#endif
  __syncthreads();

  // ---- pull this thread's 16 values into registers ----
  float v[16];
  unsigned b[16];
#pragma unroll
  for (int c = 0; c < 4; ++c) {
    const unsigned e = (unsigned)c * QUARTER + t * 4u;
    float4 f = *(const float4*)(&s_row[e]);
    v[c * 4 + 0] = f.x; v[c * 4 + 1] = f.y;
    v[c * 4 + 2] = f.z; v[c * 4 + 3] = f.w;
  }
#pragma unroll
  for (int i = 0; i < 16; ++i) b[i] = __float_as_uint(v[i]) & 0x7fffffffu;

  // ---- pass 1: radix on abs bits [30:20] (2048 bins) ----
#pragma unroll
  for (int i = 0; i < 16; ++i) atomicAdd(&s_hist[b[i] >> 20], 1u);
  __syncthreads();
  find_kth_bin<8>(s_hist, num_keep, s_scan, s_res);
  const unsigned b1 = s_res[0];
  const unsigned k2 = num_keep - s_res[1];
  __syncthreads();

  // ---- pass 2: bits [19:10] within bucket b1 (1024 bins) ----
#pragma unroll
  for (int j = 0; j < 4; ++j) s_hist[t * 4u + j] = 0u;
  __syncthreads();
#pragma unroll
  for (int i = 0; i < 16; ++i)
    if ((b[i] >> 20) == b1) atomicAdd(&s_hist[(b[i] >> 10) & 1023u], 1u);
  __syncthreads();
  find_kth_bin<4>(s_hist, k2, s_scan, s_res);
  const unsigned b2 = s_res[0];
  const unsigned k3 = k2 - s_res[1];
  const unsigned hi21 = (b1 << 10) | b2;
  __syncthreads();

  // ---- pass 3: bits [9:0] within (b1,b2) (1024 bins) ----
#pragma unroll
  for (int j = 0; j < 4; ++j) s_hist[t * 4u + j] = 0u;
  __syncthreads();
#pragma unroll
  for (int i = 0; i < 16; ++i)
    if ((b[i] >> 10) == hi21) atomicAdd(&s_hist[b[i] & 1023u], 1u);
  __syncthreads();
  find_kth_bin<4>(s_hist, k3, s_scan, s_res);
  const unsigned thr_bits = (hi21 << 10) | s_res[0];  // exact k-th smallest |x| bits

  // ---- write x*mask and mask, coalesced b128 stores ----
#pragma unroll
  for (int c = 0; c < 4; ++c) {
    const unsigned e = (unsigned)c * QUARTER + t * 4u;
    float ov[4], om[4];
#pragma unroll
    for (int j = 0; j < 4; ++j) {
      const int i = c * 4 + j;
      const float m = (b[i] >= thr_bits) ? 1.0f : 0.0f;  // uint cmp == float cmp for abs bits
      ov[j] = v[i] * m;
      om[j] = m;
    }
    *(float4*)(out_val + row_base + e)  = make_float4(ov[0], ov[1], ov[2], ov[3]);
    *(float4*)(out_mask + row_base + e) = make_float4(om[0], om[1], om[2], om[3]);
  }
}

extern "C" void kernel_launch(void* const* d_in, const int* in_sizes, int n_in,
                              void* d_out, int out_size, void* d_ws, size_t ws_size,
                              hipStream_t stream) {
  (void)n_in; (void)out_size; (void)d_ws; (void)ws_size;
  const float* x = (const float*)d_in[0];
  float* out_val = (float*)d_out;
  const size_t n = (size_t)in_sizes[0];          // 4*4096*4096
  float* out_mask = out_val + n;                 // second tuple element, flat-concatenated
  const unsigned rows = (unsigned)(n / ROW_D);   // 16384
  topk_sparsify_26611617366613<<<dim3(rows), dim3(NT), 0, stream>>>(
      x, out_val, out_mask, (unsigned)(ROW_D / 2));
}